// RoiPoolingConv_81535659147688
// MI455X (gfx1250) — compile-verified
//
#include <hip/hip_runtime.h>

// ROI bilinear pooling for MI455X (gfx1250, wave32).
// img:  (1, 100, 100, 1024) f32, NHWC, channels contiguous
// rois: (1, 512, 4) f32 holding integer x,y,w,h
// out:  (1, 512, 7, 7, 1024) f32
//
// Memory-bound target: ~103 MB NT stores + ~41 MB L2-resident image reads
// => ~6 us floor at 23.3 TB/s. Issue-side: one block per (roi, py) output
// row; ROI decode / y-coords / both f32 divisions run once per block.
// 128 threads (4 wave32 waves), 8 channels per lane (2 x b128), so the
// ~100-instr scalar setup is amortized over 2x the data vs 4 ch/lane.
// Blend uses the weighted-sum form (mul + 3 fma per component).

typedef float v4f __attribute__((ext_vector_type(4)));

#define POOL  7
#define IMG_H 100
#define IMG_W 100
#define IMG_C 1024
#define CPL   8                      // channels per lane
#define NTHR  (IMG_C / CPL)          // 128 threads = 4 waves

__global__ __launch_bounds__(NTHR) void roi_pool_bilinear(
    const float* __restrict__ img,
    const float* __restrict__ rois,
    float* __restrict__ out)
{
    const int blk = blockIdx.x;              // 0 .. R*7-1
    const int r   = blk / POOL;              // roi index
    const int py  = blk - r * POOL;          // output row 0..6

    // Uniform ROI fetch -> scalarized (s_load_b128).
    const float4 roi = reinterpret_cast<const float4*>(rois)[r];
    const int rx = (int)roi.x;
    const int ry = (int)roi.y;
    const int rw = (int)roi.z;
    const int rh = (int)roi.w;

    // ---- y direction: once per block (reference f32 chain) ----
    const float sy  = (float)py * ((float)rh / (float)POOL);
    const float fy0 = floorf(sy);
    const float fy  = sy - fy0;
    const int iy0 = (int)fy0;
    const int iy1 = min(iy0 + 1, rh - 1);
    const int ay0 = min(max(ry + iy0, 0), IMG_H - 1);
    const int ay1 = min(max(ry + iy1, 0), IMG_H - 1);

    const float* __restrict__ row0 = img + (size_t)ay0 * (IMG_W * IMG_C);
    const float* __restrict__ row1 = img + (size_t)ay1 * (IMG_W * IMG_C);

    // Single f32 division for the whole row (reference computes size/P once).
    const float scale_x = (float)rw / (float)POOL;
    const float omfy    = 1.0f - fy;

    // Per-lane channel offset: 128 lanes x 8 floats == 1024 channels.
    const int c = (int)threadIdx.x * CPL;
    float* __restrict__ obase = out + (size_t)blk * (POOL * IMG_C) + c;

#pragma unroll
    for (int px = 0; px < POOL; ++px) {
        // ---- x direction: cheap per-pixel math (no division) ----
        const float sx  = (float)px * scale_x;
        const float fx0 = floorf(sx);
        const float fx  = sx - fx0;
        const int ix0 = (int)fx0;
        const int ix1 = min(ix0 + 1, rw - 1);
        const int ax0 = min(max(rx + ix0, 0), IMG_W - 1);
        const int ax1 = min(max(rx + ix1, 0), IMG_W - 1);

        const float* s00 = row0 + ax0 * IMG_C + c;
        const float* s01 = row0 + ax1 * IMG_C + c;
        const float* s10 = row1 + ax0 * IMG_C + c;
        const float* s11 = row1 + ax1 * IMG_C + c;

        // 8 coalesced b128 gathers; the +4 halves fold into the same SGPR
        // base with an immediate byte offset. RT hint: image stays in L2.
        const v4f a00 = *(const v4f*)(s00);
        const v4f b00 = *(const v4f*)(s00 + 4);
        const v4f a01 = *(const v4f*)(s01);
        const v4f b01 = *(const v4f*)(s01 + 4);
        const v4f a10 = *(const v4f*)(s10);
        const v4f b10 = *(const v4f*)(s10 + 4);
        const v4f a11 = *(const v4f*)(s11);
        const v4f b11 = *(const v4f*)(s11 + 4);

        // Uniform bilinear weights (4 scalar muls per pixel).
        const float omfx = 1.0f - fx;
        const float w00 = omfx * omfy;
        const float w01 = fx   * omfy;
        const float w10 = omfx * fy;
        const float w11 = fx   * fy;

        // mul + 3 fma per component.
        const v4f ra = a00 * w00 + a01 * w01 + a10 * w10 + a11 * w11;
        const v4f rb = b00 * w00 + b01 * w01 + b10 * w10 + b11 * w11;

        // Write-once output stream: non-temporal b128 stores protect the
        // 41MB image's residency in the 192MB L2.
        float* d = obase + (size_t)px * IMG_C;
        __builtin_nontemporal_store(ra, (v4f*)(d));
        __builtin_nontemporal_store(rb, (v4f*)(d + 4));
    }
}

extern "C" void kernel_launch(void* const* d_in, const int* in_sizes, int n_in,
                              void* d_out, int out_size, void* d_ws, size_t ws_size,
                              hipStream_t stream) {
    const float* img  = (const float*)d_in[0];
    const float* rois = (const float*)d_in[1];
    float* out        = (float*)d_out;

    const int R = in_sizes[1] / 4;          // 512 ROIs
    const int nblocks = R * POOL;           // one block per (roi, output row)

    roi_pool_bilinear<<<dim3(nblocks), dim3(NTHR), 0, stream>>>(img, rois, out);
}